// SpikingConv3DLayer_36773509989012
// MI455X (gfx1250) — compile-verified
//
#include <hip/hip_runtime.h>

// CDNA5 / gfx1250 fused spiking-conv3d scan.
// V_WMMA_F32_16X16X4_F32 (exact fp32) for both the implicit-GEMM conv and the
// recurrent spk @ (w w^T) matmul. All WMMA operand pairs are LDS-contiguous:
// B matrices quad-interleaved [kstep][n][4] (aligned b64 loads), conv K padded
// to 72 so A taps within a lane pair are kw-adjacent (single 2-addr load).

typedef float v2f __attribute__((ext_vector_type(2)));
typedef float v8f __attribute__((ext_vector_type(8)));

#define T_STEPS 32
#define KTOT    54           // C_in*3*3*3
#define N_OUT_ELEMS 33554432 // 8*32*32*64*64

__global__ __launch_bounds__(256) void snn_fused_kernel(
    const float* __restrict__ x,      // (8,2,32,64,64)
    const float* __restrict__ w,      // (32,54)
    const float* __restrict__ beta_p, // (1,)
    const float* __restrict__ b,      // (32,)
    float* __restrict__ out,          // (8,32,32,64,64)
    float* __restrict__ loss)         // scalar accumulator (pre-zeroed)
{
    // LDS (~80 KB total -> 4 workgroups / 320 KB WGP)
    __shared__ float xbuf[34 * 360 + 8]; // slices t=-1..32 of 2cin x 10h x 18w halo (+pad)
    __shared__ float spk_lds[128 * 33];  // 128 pixels x 32 ch, padded stride 33
    __shared__ float w4[18 * 32 * 4];    // conv B, quad layout [g][c_out][q], q=kw(+pad)
    __shared__ float d4[8 * 32 * 4];     // Gram d = w w^T, quad layout [kk][c_out][q]
    __shared__ float invn[32];
    __shared__ float red[256];

    const int tid  = threadIdx.x;
    const int lane = tid & 31;
    const int wid  = tid >> 5;          // wave id = M-tile id (tile row)
    const int blk  = blockIdx.x;        // 0..255
    const int bb   = blk >> 5;          // batch 0..7
    const int tile = blk & 31;          // spatial tile (8 rows x 16 cols)
    const int h0   = (tile >> 2) * 8;
    const int w0   = (tile & 3) * 16;

    const float beta   = beta_p[0];
    const float ombeta = 1.0f - beta;

    // ---- conv B: w4[g][n][q] = w[n, cin*27+kd*9+kh*3+q], g=cin*9+kd*3+kh ----
    for (int idx = tid; idx < 18 * 32 * 4; idx += 256) {
        int g = idx >> 7, rem = idx & 127;
        int n = rem >> 2, q = rem & 3;
        int cin = g / 9, r = g % 9, kd = r / 3, kh = r % 3;
        w4[idx] = (q < 3) ? w[n * KTOT + cin * 27 + kd * 9 + kh * 3 + q] : 0.0f;
    }
    // ---- zero spike tile (spk(t=-1) == 0) ----
    for (int idx = tid; idx < 128 * 33; idx += 256) spk_lds[idx] = 0.0f;
    // ---- load the full x halo tile: slice s lives at (s+1)*360 ----
    for (int s = -1; s <= 32; ++s) {
        for (int idx = tid; idx < 360; idx += 256) {
            int cin = idx / 180, rem = idx % 180;
            int hh = rem / 18, ww = rem % 18;
            int h = h0 - 1 + hh, wq = w0 - 1 + ww;
            float v = 0.0f;
            if (s >= 0 && s < 32 && h >= 0 && h < 64 && wq >= 0 && wq < 64)
                v = x[(((bb * 2 + cin) * 32 + s) << 12) + (h << 6) + wq];
            xbuf[(s + 1) * 360 + idx] = v;
        }
    }
    if (tid < 8) xbuf[34 * 360 + tid] = 0.0f;
    __syncthreads();

    // ---- Gram matrix into quad layout: d4[i>>2][j][i&3] = sum_k w[i,k] w[j,k] ----
    for (int e = 0; e < 4; ++e) {
        int idx = tid * 4 + e;
        int i = idx >> 5, j = idx & 31;       // d[i][j]
        float s = 0.0f;
        for (int gq = 0; gq < 72; ++gq)       // pad slots are zero in w4
            s += w4[(gq >> 2) * 128 + i * 4 + (gq & 3)] *
                 w4[(gq >> 2) * 128 + j * 4 + (gq & 3)];
        d4[(i >> 2) * 128 + j * 4 + (i & 3)] = s;
    }
    if (tid < 32) {
        float s = 0.0f;
        for (int gq = 0; gq < 72; ++gq) {
            float v = w4[(gq >> 2) * 128 + tid * 4 + (gq & 3)];
            s += v * v;
        }
        invn[tid] = 1.0f / (s + 1e-8f);
    }
    __syncthreads();

    // ---- per-lane constants (WMMA lane mapping) ----
    const int arow   = lane & 15;            // A row within M-tile; also N%16
    const int khalf  = (lane >> 4) * 2;      // lanes 16-31 hold K = 4kk+{2,3}
    const int pixoff = wid * 18 + arow;      // ph*18 + pw into a halo slice
    const int pA     = wid * 16 + arow;      // A-row pixel index (0..127)
    const int sbase  = pA * 33 + khalf;      // rst A base (t-invariant)
    const int wqb    = arow * 2 + (khalf >> 1);  // v2f index into w4/d4 quads
    const float in0  = invn[arow],      in1 = invn[arow + 16];
    const float bi0  = b[arow],         bi1 = b[arow + 16];

    const v2f* w4v = (const v2f*)w4;
    const v2f* d4v = (const v2f*)d4;

    // per-lane conv A bases: addr(t, g) = t*360 + gbase[g] (+0/+1 within pair)
    int gbase[18];
#pragma unroll
    for (int g = 0; g < 18; ++g) {
        int cin = g / 9, r = g % 9, kd = r / 3, kh = r % 3;
        gbase[g] = kd * 360 + cin * 180 + kh * 18 + pixoff + khalf;
    }

    // per-thread invariant pieces of the output-store addressing
    const int pS  = tid & 127;                          // staging pixel
    const int cS0 = tid >> 7;                           // first channel (0 or 1)
    const int pixglob = ((h0 + (pS >> 4)) << 6) + (w0 + (pS & 15));
    const int outbase0 = bb * (32 * 32 * 4096) + cS0 * (32 * 4096) + pixglob;

    v8f mem0 = {0.f,0.f,0.f,0.f,0.f,0.f,0.f,0.f};
    v8f mem1 = {0.f,0.f,0.f,0.f,0.f,0.f,0.f,0.f};
    float lsum = 0.0f;

    for (int t = 0; t < T_STEPS; ++t) {
        const int tb = t * 360;

        // ---- conv implicit GEMM: inp = patch(128x72) @ w(72x32) ----
        v8f inp0 = {0.f,0.f,0.f,0.f,0.f,0.f,0.f,0.f};
        v8f inp1 = {0.f,0.f,0.f,0.f,0.f,0.f,0.f,0.f};
#pragma unroll
        for (int g = 0; g < 18; ++g) {
            int ab = tb + gbase[g];
            v2f a;  a.x = xbuf[ab];  a.y = xbuf[ab + 1];   // kw-adjacent pair
            v2f b0 = w4v[g * 64 + wqb];                    // aligned b64
            v2f b1 = w4v[g * 64 + wqb + 32];               // n + 16
            inp0 = __builtin_amdgcn_wmma_f32_16x16x4_f32(false, a, false, b0, (short)0, inp0, false, false);
            inp1 = __builtin_amdgcn_wmma_f32_16x16x4_f32(false, a, false, b1, (short)0, inp1, false, false);
        }

        // ---- recurrent GEMM: rst = spk(128x32) @ d(32x32) ----
        v8f rst0 = {0.f,0.f,0.f,0.f,0.f,0.f,0.f,0.f};
        v8f rst1 = {0.f,0.f,0.f,0.f,0.f,0.f,0.f,0.f};
#pragma unroll
        for (int kk = 0; kk < 8; ++kk) {
            v2f a;  a.x = spk_lds[sbase + 4 * kk];  a.y = spk_lds[sbase + 4 * kk + 1];
            v2f b0 = d4v[kk * 64 + wqb];
            v2f b1 = d4v[kk * 64 + wqb + 32];
            rst0 = __builtin_amdgcn_wmma_f32_16x16x4_f32(false, a, false, b0, (short)0, rst0, false, false);
            rst1 = __builtin_amdgcn_wmma_f32_16x16x4_f32(false, a, false, b1, (short)0, rst1, false, false);
        }

        // ---- leaky integrate + threshold (registers) ----
        v8f spk0, spk1;
#pragma unroll
        for (int r = 0; r < 8; ++r) {
            mem0[r] = (mem0[r] - rst0[r]) * beta + inp0[r] * ombeta;
            mem1[r] = (mem1[r] - rst1[r]) * beta + inp1[r] * ombeta;
            spk0[r] = (mem0[r] * in0 - bi0) > 0.0f ? 1.0f : 0.0f;
            spk1[r] = (mem1[r] * in1 - bi1) > 0.0f ? 1.0f : 0.0f;
        }

        __syncthreads();                      // all waves done reading old spikes
        {   // D-layout scatter into padded LDS tile
            int Mbase = (lane >> 4) * 8;
#pragma unroll
            for (int r = 0; r < 8; ++r) {
                int p = wid * 16 + r + Mbase;
                spk_lds[p * 33 + arow]      = spk0[r];
                spk_lds[p * 33 + arow + 16] = spk1[r];
            }
        }
        __syncthreads();

        // ---- coalesced global store of spk_rec[b, c, t, h, w] + spike count ----
        {
            const float* sp = &spk_lds[pS * 33 + cS0];
            float* op = out + outbase0 + (t << 12);
#pragma unroll 4
            for (int i = 0; i < 16; ++i) {
                float v = sp[2 * i];
                lsum += v;
                op[i * (2 * 32 * 4096)] = v;
            }
        }
    }

    // ---- loss: 0.5 * mean(spk^2) = 0.5 * sum(spk)/N, spk in {0,1} ----
    red[tid] = lsum;
    __syncthreads();
    for (int s = 128; s > 0; s >>= 1) {
        if (tid < s) red[tid] += red[tid + s];
        __syncthreads();
    }
    if (tid == 0) atomicAdd(loss, red[0] * (0.5f / (float)N_OUT_ELEMS));
}

extern "C" void kernel_launch(void* const* d_in, const int* in_sizes, int n_in,
                              void* d_out, int out_size, void* d_ws, size_t ws_size,
                              hipStream_t stream) {
    (void)in_sizes; (void)n_in; (void)out_size; (void)d_ws; (void)ws_size;
    const float* x    = (const float*)d_in[0];
    const float* w    = (const float*)d_in[1];
    const float* beta = (const float*)d_in[2];
    const float* b    = (const float*)d_in[3];
    float* out  = (float*)d_out;
    float* loss = out + N_OUT_ELEMS;
    hipMemsetAsync(loss, 0, sizeof(float), stream);   // graph-capture safe
    snn_fused_kernel<<<dim3(256), dim3(256), 0, stream>>>(x, w, beta, b, out, loss);
}